// SegmentEmbedding_90280212562591
// MI455X (gfx1250) — compile-verified
//
#include <hip/hip_runtime.h>

typedef __attribute__((ext_vector_type(2))) float v2f;
typedef __attribute__((ext_vector_type(8))) float v8f;

#define BB 32
#define LL 4096
#define DD 768

// ---------------- Phase 1: find SEP (==4) and EOS (==2) positions per row ---
__global__ void seg_markers_kernel(const long long* __restrict__ x,
                                   int* __restrict__ sepeos) {
  __shared__ int s_sep, s_eos;
  const int b = blockIdx.x;
  if (threadIdx.x == 0) { s_sep = 0x7fffffff; s_eos = 0x7fffffff; }
  __syncthreads();
  int lsep = 0x7fffffff, leos = 0x7fffffff;
  const long long* row = x + (size_t)b * LL;
  for (int i = threadIdx.x; i < LL; i += blockDim.x) {
    long long v = row[i];
    if (v == 4 && i < lsep) lsep = i;
    if (v == 2 && i < leos) leos = i;
  }
  atomicMin(&s_sep, lsep);
  atomicMin(&s_eos, leos);
  __syncthreads();
  if (threadIdx.x == 0) { sepeos[b] = s_sep; sepeos[BB + b] = s_eos; }
}

// ---------------- Phase 2: out = onehot(seg) x W via V_WMMA_F32_16X16X4_F32 --
// Wave decomposition: global wave g -> (chunk of 16 positions) x (192 columns).
// A (16x4 f32, one-hot of seg):  VGPR0 = K=0 (lanes 0-15) / K=2 (lanes 16-31),
//                                VGPR1 = K=1 / K=3.
// B (4x16 f32, rows of W):       same half-wave striping; row 3 is zero.
// D (16x16 f32): VGPR k -> row k (lanes 0-15) / row k+8 (lanes 16-31), col = lane%16.
__global__ void __launch_bounds__(256)
seg_embed_wmma_kernel(const float* __restrict__ W,
                      const int* __restrict__ sepeos,
                      float* __restrict__ out) {
  __shared__ float sW[3 * DD];
  for (int i = threadIdx.x; i < 3 * DD; i += 256) sW[i] = W[i];
  __syncthreads();

  const int lane = threadIdx.x & 31;
  const int wib  = __builtin_amdgcn_readfirstlane((int)(threadIdx.x >> 5));
  const int g    = blockIdx.x * 8 + wib;     // global wave id: 0..32767
  const int colgroup = g & 3;                // 4 groups of 192 columns
  const int chunk    = g >> 2;               // 0..8191 (16-position chunks)
  const int b        = chunk >> 8;           // 256 chunks per row
  const int posbase  = (chunk & 255) << 4;

  const int sep = sepeos[b];                 // wave-uniform -> scalar loads
  const int eos = sepeos[BB + b];

  const int M  = lane & 15;                  // tile row (position) / tile col
  const int hi = lane >> 4;
  const int k0 = hi << 1;                    // this half-wave holds K rows {k0, k0+1}

  // one-hot A fragment from segment id of position posbase+M
  const int pos = posbase + M;
  const int seg = (pos < sep) ? 1 : ((pos < eos) ? 2 : 0);
  v2f a;
  a.x = (seg == k0)     ? 1.0f : 0.0f;
  a.y = (seg == k0 + 1) ? 1.0f : 0.0f;       // seg==3 never happens -> row 3 unused

  const int cg0 = colgroup * 192;
  float* const orow = out + (size_t)(b * LL + posbase + (hi << 3)) * DD;

  #pragma unroll
  for (int t = 0; t < 12; ++t) {
    const int col = cg0 + t * 16 + M;
    v2f bm;
    bm.x = sW[k0 * DD + col];                // W row k0 (0 or 2)
    float w1 = sW[DD + col];                 // W row 1
    bm.y = hi ? 0.0f : w1;                   // K=3 row of B is zero
    v8f acc = {0.f, 0.f, 0.f, 0.f, 0.f, 0.f, 0.f, 0.f};
    acc = __builtin_amdgcn_wmma_f32_16x16x4_f32(
        /*neg_a=*/false, a, /*neg_b=*/false, bm,
        /*c_mod=*/(short)0, acc, /*reuse_a=*/false, /*reuse_b=*/false);
    float* p = orow + col;
    #pragma unroll
    for (int k = 0; k < 8; ++k)
      __builtin_nontemporal_store(acc[k], p + (size_t)k * DD);  // streaming, NT
  }
}

extern "C" void kernel_launch(void* const* d_in, const int* in_sizes, int n_in,
                              void* d_out, int out_size, void* d_ws, size_t ws_size,
                              hipStream_t stream) {
  const long long* x = (const long long*)d_in[0];   // int64 tokens [32,4096]
  const float*     W = (const float*)d_in[1];       // [3,768] f32
  float*         out = (float*)d_out;               // [32,4096,768] f32
  int*        sepeos = (int*)d_ws;                  // 64 ints of scratch

  seg_markers_kernel<<<BB, 256, 0, stream>>>(x, sepeos);

  const int totalWaves = (BB * LL / 16) * 4;        // 32768 waves, 8 per block
  seg_embed_wmma_kernel<<<totalWaves / 8, 256, 0, stream>>>(W, sepeos, out);
}